// ResidualMambaBlock_30434138260197
// MI455X (gfx1250) — compile-verified
//
#include <hip/hip_runtime.h>
#include <math.h>
#include <stdint.h>

#define B_    2
#define L_    2048
#define D_    768
#define E_    (2 * D_)        // 1536
#define N_    16
#define K_    4
#define R_    (D_ / 16)       // 48
#define XDBL_ (R_ + 2 * N_)   // 80
#define ROWS_ (B_ * L_)       // 4096
#define EPS_  1e-6f

typedef __attribute__((ext_vector_type(2))) float v2f;
typedef __attribute__((ext_vector_type(8))) float v8f;

// ---------------------------------------------------------------------------
// CDNA5 async global->LDS copy (ASYNCcnt-tracked), per CDNA5 ISA.
// ---------------------------------------------------------------------------
__device__ __forceinline__ void async_copy_b128(unsigned lds_off, const float* g) {
    asm volatile("global_load_async_to_lds_b128 %0, %1, off"
                 :: "v"(lds_off), "v"(g) : "memory");
}
__device__ __forceinline__ void wait_async_le3() {
    asm volatile("s_wait_asynccnt 0x3" ::: "memory");
}
__device__ __forceinline__ void wait_async_le0() {
    asm volatile("s_wait_asynccnt 0x0" ::: "memory");
}
__device__ __forceinline__ unsigned lds_off_of(const void* p) {
    // low 32 bits of the flat shared-aperture address = LDS byte offset
    return (unsigned)(uintptr_t)p;
}

// ---------------------------------------------------------------------------
// RMSNorm: one block per (b,l) row over D=768
// ---------------------------------------------------------------------------
__global__ void mamba_rmsnorm_kernel(const float* __restrict__ x,
                                     const float* __restrict__ w,
                                     float* __restrict__ h) {
    const int row = blockIdx.x;
    const float* xr = x + (size_t)row * D_;
    float* hr = h + (size_t)row * D_;

    float ss = 0.f;
    for (int i = threadIdx.x; i < D_; i += blockDim.x) {
        float v = xr[i];
        ss += v * v;
    }
    __shared__ float red[8];
    #pragma unroll
    for (int off = 16; off > 0; off >>= 1) ss += __shfl_down(ss, off, 32);
    const int wave = threadIdx.x >> 5;
    const int lane = threadIdx.x & 31;
    if (lane == 0) red[wave] = ss;
    __syncthreads();
    if (wave == 0) {
        float v = (lane < 8) ? red[lane] : 0.f;
        #pragma unroll
        for (int off = 4; off > 0; off >>= 1) v += __shfl_down(v, off, 32);
        if (lane == 0) red[0] = v;
    }
    __syncthreads();
    const float scale = rsqrtf(red[0] / (float)D_ + EPS_);
    for (int i = threadIdx.x; i < D_; i += blockDim.x)
        hr[i] = xr[i] * scale * w[i];
}

// ---------------------------------------------------------------------------
// Tiled f32 WMMA GEMM with double-buffered async LDS staging.
// Block = 256 threads (8 waves) -> 64x128 macro-tile; wave = 16x64 (4 tiles).
// Requires M%64==0, N%128==0, K%16==0.
// MODE 0: plain   MODE 1: softplus(v + ex[col])   MODE 2: v + res[row,col]
// ---------------------------------------------------------------------------
template <int MODE>
__global__ void mamba_gemm_wmma_tiled(const float* __restrict__ A,
                                      const float* __restrict__ Bm,
                                      float* __restrict__ C,
                                      const float* __restrict__ ex,
                                      const float* __restrict__ res,
                                      int Kd, int lda, int ldb, int ldc, int nb) {
    __shared__ float As[2][64][20];    // A panel 64x16, padded row (bank spread)
    __shared__ float Bs[2][16][128];   // B panel 16x128

    const int bm = blockIdx.x / nb;
    const int bn = blockIdx.x - bm * nb;
    const int wid  = threadIdx.x >> 5;
    const int lane = threadIdx.x & 31;
    const int wm = wid & 3;            // wave's M tile (0..3)
    const int wn = wid >> 2;           // wave's N group (0..1)
    const int half = lane >> 4;        // K-pair select inside a WMMA step
    const int l16  = lane & 15;

    // ---- async-copy source/dest assignments (per thread) ----
    const int tA_r = threadIdx.x >> 2;          // 0..63
    const int tA_c = (threadIdx.x & 3) << 2;    // 0,4,8,12
    const float* aSrc = A + (size_t)(bm * 64 + tA_r) * lda + tA_c;
    unsigned asOff[2];
    asOff[0] = lds_off_of(&As[0][tA_r][tA_c]);
    asOff[1] = lds_off_of(&As[1][tA_r][tA_c]);

    const int v0 = threadIdx.x;                 // B vec4 slot 0
    const int v1 = threadIdx.x + 256;           // B vec4 slot 1
    const int bR0 = v0 >> 5, bC0 = (v0 & 31) << 2;
    const int bR1 = v1 >> 5, bC1 = (v1 & 31) << 2;
    const float* bSrc0 = Bm + (size_t)bR0 * ldb + bn * 128 + bC0;
    const float* bSrc1 = Bm + (size_t)bR1 * ldb + bn * 128 + bC1;
    unsigned bsOff0[2], bsOff1[2];
    bsOff0[0] = lds_off_of(&Bs[0][bR0][bC0]);
    bsOff0[1] = lds_off_of(&Bs[1][bR0][bC0]);
    bsOff1[0] = lds_off_of(&Bs[0][bR1][bC1]);
    bsOff1[1] = lds_off_of(&Bs[1][bR1][bC1]);

    auto issue_chunk = [&](int kc) {
        const int buf = kc & 1;
        const int k0 = kc << 4;
        async_copy_b128(asOff[buf],  aSrc + k0);
        async_copy_b128(bsOff0[buf], bSrc0 + (size_t)k0 * ldb);
        async_copy_b128(bsOff1[buf], bSrc1 + (size_t)k0 * ldb);
    };

    v8f acc[4] = {};
    const int nk = Kd >> 4;
    issue_chunk(0);
    for (int kc = 0; kc < nk; ++kc) {
        if (kc + 1 < nk) { issue_chunk(kc + 1); wait_async_le3(); }
        else             { wait_async_le0(); }
        __syncthreads();
        const int buf = kc & 1;
        #pragma unroll
        for (int kk = 0; kk < 4; ++kk) {
            const int kb = (kk << 2) + (half << 1);
            v2f a;
            a.x = As[buf][wm * 16 + l16][kb];
            a.y = As[buf][wm * 16 + l16][kb + 1];
            #pragma unroll
            for (int j = 0; j < 4; ++j) {
                v2f b;
                b.x = Bs[buf][kb][wn * 64 + j * 16 + l16];
                b.y = Bs[buf][kb + 1][wn * 64 + j * 16 + l16];
                acc[j] = __builtin_amdgcn_wmma_f32_16x16x4_f32(
                    false, a, false, b, (short)0, acc[j], false, false);
            }
        }
        __syncthreads();
    }

    // ---- epilogue: 16x64 per wave ----
    const int colBase = bn * 128 + wn * 64;
    #pragma unroll
    for (int j = 0; j < 4; ++j) {
        const int col = colBase + j * 16 + l16;
        #pragma unroll
        for (int r = 0; r < 8; ++r) {
            const int row = bm * 64 + wm * 16 + r + (half << 3);
            float v = acc[j][r];
            if (MODE == 1) {                       // softplus(v + dt_bias[col])
                v += ex[col];
                v = (v > 20.f) ? v : log1pf(__expf(v));
            } else if (MODE == 2) {                // residual add
                v += res[(size_t)row * ldc + col];
            }
            C[(size_t)row * ldc + col] = v;
        }
    }
}

// ---------------------------------------------------------------------------
// Simple f32 WMMA GEMM (wave = one 16x16 tile), for N not divisible by 128.
// ---------------------------------------------------------------------------
__global__ void mamba_gemm_wmma_small(const float* __restrict__ A,
                                      const float* __restrict__ Bm,
                                      float* __restrict__ C,
                                      int M, int N, int Kd,
                                      int lda, int ldb, int ldc) {
    const int tilesN = N >> 4;
    const int totalTiles = (M >> 4) * tilesN;
    const int tile = blockIdx.x * (blockDim.x >> 5) + (threadIdx.x >> 5);
    if (tile >= totalTiles) return;   // wave-uniform: EXEC stays all-ones

    const int tm = tile / tilesN;
    const int tn = tile - tm * tilesN;
    const int lane = threadIdx.x & 31;
    const int half = lane >> 4;
    const int l16  = lane & 15;

    const float* Arow = A + (size_t)(tm * 16 + l16) * lda;
    const float* Bcol = Bm + (tn * 16 + l16);

    v8f acc = {};
    for (int k0 = 0; k0 < Kd; k0 += 4) {
        const int koff = k0 + (half << 1);
        v2f a, b;
        a.x = Arow[koff];
        a.y = Arow[koff + 1];
        b.x = Bcol[(size_t)koff * ldb];
        b.y = Bcol[(size_t)(koff + 1) * ldb];
        acc = __builtin_amdgcn_wmma_f32_16x16x4_f32(
            false, a, false, b, (short)0, acc, false, false);
    }
    const int col = tn * 16 + l16;
    #pragma unroll
    for (int r = 0; r < 8; ++r) {
        const int row = tm * 16 + r + (half << 3);
        C[(size_t)row * ldc + col] = acc[r];
    }
}

// ---------------------------------------------------------------------------
// Causal depthwise conv1d (K=4) + SiLU. xm is first E cols of xz rows.
// ---------------------------------------------------------------------------
__global__ void mamba_conv_silu_kernel(const float* __restrict__ xz,
                                       const float* __restrict__ cw,
                                       const float* __restrict__ cb,
                                       float* __restrict__ xc) {
    const int idx = blockIdx.x * blockDim.x + threadIdx.x;   // row*E + e
    if (idx >= ROWS_ * E_) return;
    const int e = idx % E_;
    const int row = idx / E_;
    const int l = row % L_;

    float acc = cb[e];
    #pragma unroll
    for (int k = 0; k < K_; ++k) {
        const int lk = l - (K_ - 1) + k;
        if (lk >= 0)
            acc += xz[(size_t)(row - (K_ - 1) + k) * (2 * E_) + e] * cw[e * K_ + k];
    }
    xc[idx] = acc / (1.f + __expf(-acc));    // silu
}

// ---------------------------------------------------------------------------
// Selective scan: lane = one state n of one channel (b,e).
// 16-lane shfl_xor reduction for y_l = sum_n h_n C_n; gating fused.
// ---------------------------------------------------------------------------
__global__ void mamba_scan_kernel(const float* __restrict__ dt,
                                  const float* __restrict__ xdbl,
                                  const float* __restrict__ xc,
                                  const float* __restrict__ xz,
                                  const float* __restrict__ A_log,
                                  const float* __restrict__ Dskip,
                                  float* __restrict__ y) {
    const int gid = blockIdx.x * blockDim.x + threadIdx.x;
    const int c = gid >> 4;              // channel index over B*E
    if (c >= B_ * E_) return;
    const int n = gid & 15;
    const int b = c / E_;
    const int e = c - b * E_;

    const float An  = -__expf(A_log[e * N_ + n]);
    const float dsk = Dskip[e];
    float hn = 0.f;

    for (int l = 0; l < L_; ++l) {
        const size_t row = (size_t)b * L_ + l;
        const float dt_v = dt[row * E_ + e];
        const float x_v  = xc[row * E_ + e];
        const float Bn   = xdbl[row * XDBL_ + R_ + n];
        const float Cn   = xdbl[row * XDBL_ + R_ + N_ + n];

        // hide latency of the serial chain with prefetches a few steps ahead
        __builtin_prefetch(&dt[(row + 8) * E_ + e], 0, 0);
        __builtin_prefetch(&xdbl[(row + 8) * XDBL_ + R_ + n], 0, 0);

        hn = __expf(dt_v * An) * hn + dt_v * x_v * Bn;
        float p = hn * Cn;
        p += __shfl_xor(p, 1, 32);
        p += __shfl_xor(p, 2, 32);
        p += __shfl_xor(p, 4, 32);
        p += __shfl_xor(p, 8, 32);       // all 16 lanes of the group hold sum
        if (n == 0) {
            const float z_v = xz[row * (2 * E_) + E_ + e];
            y[row * E_ + e] = (p + dsk * x_v) * (z_v / (1.f + __expf(-z_v)));
        }
    }
}

// ---------------------------------------------------------------------------
extern "C" void kernel_launch(void* const* d_in, const int* in_sizes, int n_in,
                              void* d_out, int out_size, void* d_ws, size_t ws_size,
                              hipStream_t stream) {
    (void)in_sizes; (void)n_in; (void)out_size; (void)ws_size;

    const float* x      = (const float*)d_in[0];
    const float* norm_w = (const float*)d_in[1];
    const float* W_in   = (const float*)d_in[2];
    const float* conv_w = (const float*)d_in[3];
    const float* conv_b = (const float*)d_in[4];
    const float* W_x    = (const float*)d_in[5];
    const float* W_dt   = (const float*)d_in[6];
    const float* dt_b   = (const float*)d_in[7];
    const float* A_log  = (const float*)d_in[8];
    const float* D_skip = (const float*)d_in[9];
    const float* W_out  = (const float*)d_in[10];
    float* out = (float*)d_out;

    float* ws = (float*)d_ws;
    size_t o = 0;
    float* h    = ws + o; o += (size_t)ROWS_ * D_;
    float* xz   = ws + o; o += (size_t)ROWS_ * 2 * E_;
    float* xc   = ws + o; o += (size_t)ROWS_ * E_;
    float* xdbl = ws + o; o += (size_t)ROWS_ * XDBL_;
    float* dt   = ws + o; o += (size_t)ROWS_ * E_;
    float* y    = ws + o;

    // 1) RMSNorm
    mamba_rmsnorm_kernel<<<ROWS_, 256, 0, stream>>>(x, norm_w, h);

    // 2) xz = h @ W_in   [4096,768]x[768,3072]  (tiled, async LDS staging)
    {
        const int nb = (2 * E_) / 128;                 // 24
        mamba_gemm_wmma_tiled<0><<<(ROWS_ / 64) * nb, 256, 0, stream>>>(
            h, W_in, xz, nullptr, nullptr, D_, D_, 2 * E_, 2 * E_, nb);
    }

    // 3) causal depthwise conv + SiLU -> xc
    mamba_conv_silu_kernel<<<(ROWS_ * E_ + 255) / 256, 256, 0, stream>>>(
        xz, conv_w, conv_b, xc);

    // 4) xdbl = xc @ W_x   [4096,1536]x[1536,80]  (N=80 -> simple path)
    {
        int tiles = (ROWS_ / 16) * (XDBL_ / 16);
        mamba_gemm_wmma_small<<<(tiles + 7) / 8, 256, 0, stream>>>(
            xc, W_x, xdbl, ROWS_, XDBL_, E_, E_, XDBL_, XDBL_);
    }

    // 5) dt = softplus(xdbl[:, :48] @ W_dt + dt_bias)   [4096,48]x[48,1536]
    {
        const int nb = E_ / 128;                       // 12
        mamba_gemm_wmma_tiled<1><<<(ROWS_ / 64) * nb, 256, 0, stream>>>(
            xdbl, W_dt, dt, dt_b, nullptr, R_, XDBL_, E_, E_, nb);
    }

    // 6) selective scan + gating -> y
    mamba_scan_kernel<<<(B_ * E_ * 16 + 255) / 256, 256, 0, stream>>>(
        dt, xdbl, xc, xz, A_log, D_skip, y);

    // 7) out = y @ W_out + x   [4096,1536]x[1536,768] with residual
    {
        const int nb = D_ / 128;                       // 6
        mamba_gemm_wmma_tiled<2><<<(ROWS_ / 64) * nb, 256, 0, stream>>>(
            y, W_out, out, nullptr, x, E_, E_, D_, D_, nb);
    }
}